// NeuralSort_8778913153218
// MI455X (gfx1250) — compile-verified
//
#include <hip/hip_runtime.h>

// NeuralSort forward: b=64, n=1024, f32 in/out.
// out[b,i] = sum_j softmax_j(scaling_i*s_j - rowsum_j) * s_j
// scaling_i = n-1-2i ; rowsum_j = sum_k |s_j - s_k|

typedef __attribute__((ext_vector_type(2))) float v2f;
typedef __attribute__((ext_vector_type(8))) float v8f;

#define NS_N 1024
#define NS_B 64

// ---------------------------------------------------------------------------
// Kernel 1: rowsum[b][j] = sum_k |s[b][j] - s[b][k]|
// One block per batch row, scores staged in LDS, 4 j's per thread.
// ---------------------------------------------------------------------------
__global__ __launch_bounds__(256) void ns_rowsum_kernel(
    const float* __restrict__ scores, float* __restrict__ rowsum) {
  __shared__ __align__(16) float s[NS_N];
  const int b = blockIdx.x;
  const int t = threadIdx.x;
  const float* sb = scores + b * NS_N;

  #pragma unroll
  for (int j = t; j < NS_N; j += 256) s[j] = sb[j];
  __syncthreads();

  const int j0 = t * 4;
  const float sj0 = s[j0 + 0], sj1 = s[j0 + 1];
  const float sj2 = s[j0 + 2], sj3 = s[j0 + 3];
  float a0 = 0.f, a1 = 0.f, a2 = 0.f, a3 = 0.f;

  for (int k = 0; k < NS_N; k += 4) {
    const float4 sk = *(const float4*)&s[k];
    a0 += fabsf(sj0 - sk.x) + fabsf(sj0 - sk.y) + fabsf(sj0 - sk.z) + fabsf(sj0 - sk.w);
    a1 += fabsf(sj1 - sk.x) + fabsf(sj1 - sk.y) + fabsf(sj1 - sk.z) + fabsf(sj1 - sk.w);
    a2 += fabsf(sj2 - sk.x) + fabsf(sj2 - sk.y) + fabsf(sj2 - sk.z) + fabsf(sj2 - sk.w);
    a3 += fabsf(sj3 - sk.x) + fabsf(sj3 - sk.y) + fabsf(sj3 - sk.z) + fabsf(sj3 - sk.w);
  }
  float* rb = rowsum + b * NS_N + j0;
  rb[0] = a0; rb[1] = a1; rb[2] = a2; rb[3] = a3;
}

// ---------------------------------------------------------------------------
// Kernel 2: softmax + weighted sum.
// grid (64, 8), block 256 = 8 wave32s. Wave w of blockIdx.y=g owns the 16-row
// i-tile  T = g*8 + w  (64 tiles cover i = 0..1023 for batch blockIdx.x).
//
// Inner loop steps j by 4 and feeds V_WMMA_F32_16X16X4_F32:
//   A[row, k] = exp(scaling_row * s_{j+k} - rowsum_{j+k} - max_row)
//   B[k, 0]   = s_{j+k},  B[k, 1] = 1.0,  other cols 0
// => D[:,0] accumulates the numerator, D[:,1] the denominator.
// ---------------------------------------------------------------------------
__global__ __launch_bounds__(256) void ns_softmax_kernel(
    const float* __restrict__ scores, const float* __restrict__ rowsum,
    float* __restrict__ out) {
  __shared__ __align__(16) float s[NS_N];
  __shared__ __align__(16) float rs[NS_N];
  __shared__ __align__(16) float stage[8][32];  // per wave: [0..15]=num, [16..31]=den

  const int b    = blockIdx.x;
  const int t    = threadIdx.x;
  const int wave = t >> 5;
  const int lane = t & 31;

  // cooperative stage of this batch row (uniform trip count -> EXEC stays full)
  #pragma unroll
  for (int j = t; j < NS_N; j += 256) {
    s[j]  = scores[b * NS_N + j];
    rs[j] = rowsum[b * NS_N + j];
  }
  __syncthreads();

  const int tile = blockIdx.y * 8 + wave;   // 0..63
  const int i0   = tile * 16;
  const int row  = lane & 15;               // M (for A) / N (for B) coordinate
  const bool hi  = lane >= 16;
  const int kofs = hi ? 2 : 0;              // K offset this lane supplies for A/B

  // scaling for this lane's matrix row (same for lane l and l+16)
  const float c = (float)(NS_N - 1 - 2 * (i0 + row));

  // ---- pass 1: per-row max of (c*s_j - rs_j); lanes split j-range in half ----
  float m = -3.0e38f;
  const int jb = hi ? (NS_N / 2) : 0;
  for (int j = jb; j < jb + NS_N / 2; j += 4) {
    const float4 sv = *(const float4*)&s[j];
    const float4 rv = *(const float4*)&rs[j];
    m = fmaxf(m, fmaf(c, sv.x, -rv.x));
    m = fmaxf(m, fmaf(c, sv.y, -rv.y));
    m = fmaxf(m, fmaf(c, sv.z, -rv.z));
    m = fmaxf(m, fmaf(c, sv.w, -rv.w));
  }
  m = fmaxf(m, __shfl_xor(m, 16, 32));      // now every lane has max of its row

  // ---- pass 2: exp on VALU, num/den reduction on the WMMA pipe ----
  const float bfill = (row == 1) ? 1.0f : 0.0f;  // column-1 = ones, cols>=2 = 0
  v8f acc = {};
  for (int j = 0; j < NS_N; j += 4) {
    const float2 sv = *(const float2*)&s[j + kofs];   // 8B aligned (kofs in {0,2})
    const float2 rv = *(const float2*)&rs[j + kofs];
    const float a0 = __expf(fmaf(c, sv.x, -rv.x) - m);
    const float a1 = __expf(fmaf(c, sv.y, -rv.y) - m);
    v2f A = {a0, a1};
    v2f Bm = {(row == 0) ? sv.x : bfill,
              (row == 0) ? sv.y : bfill};
    acc = __builtin_amdgcn_wmma_f32_16x16x4_f32(
        /*neg_a=*/false, A, /*neg_b=*/false, Bm,
        /*c_mod=*/(short)0, acc, /*reuse_a=*/false, /*reuse_b=*/false);
  }

  // ---- writeback: D layout = VGPR d holds row d (lanes 0-15) / row d+8 ----
  if (row < 2) {  // lanes 0,1,16,17 carry D columns 0 (num) and 1 (den)
    const int base = row * 16 + (hi ? 8 : 0);
    #pragma unroll
    for (int d = 0; d < 8; ++d) stage[wave][base + d] = acc[d];
  }
  __syncthreads();

  if (lane < 16)
    out[b * NS_N + i0 + lane] = stage[wave][lane] / stage[wave][16 + lane];
}

// ---------------------------------------------------------------------------
extern "C" void kernel_launch(void* const* d_in, const int* in_sizes, int n_in,
                              void* d_out, int out_size, void* d_ws, size_t ws_size,
                              hipStream_t stream) {
  (void)in_sizes; (void)n_in; (void)out_size; (void)ws_size;
  const float* scores = (const float*)d_in[0];
  float* out    = (float*)d_out;
  float* rowsum = (float*)d_ws;               // 64*1024 floats = 256 KB scratch

  ns_rowsum_kernel<<<dim3(NS_B), 256, 0, stream>>>(scores, rowsum);
  ns_softmax_kernel<<<dim3(NS_B, 8), 256, 0, stream>>>(scores, rowsum, out);
}